// GNNModel_34179349742294
// MI455X (gfx1250) — compile-verified
//
#include <hip/hip_runtime.h>
#include <hip/hip_bf16.h>

// ---------------------------------------------------------------------------
// GNN: two GraphConv layers (sym-normalized SpMM + dense linear) on gfx1250.
//  * SpMM (edge gather + scatter-add) dominates: ~3.3 GB gather + ~3.3 GB
//    atomic traffic per layer; working set ~205 MB ~= global L2 (192 MB),
//    so most of it is on-chip. Non-returning f32 atomics (STOREcnt path).
//  * GEMMs run on FP32 WMMA (v_wmma_f32_16x16x4_f32) -- fp32 fidelity at
//    zero perf cost since GEMM flops/bytes are small next to the SpMM.
//    Each wave computes a 16x64 strip (4 accumulators) so one A-fragment
//    feeds 4 WMMAs (4x less A traffic); the 256x64 weight panel is staged
//    in LDS (64 KB) per workgroup via GLOBAL_LOAD_ASYNC_TO_LDS_B128.
// ---------------------------------------------------------------------------

typedef float v2f __attribute__((ext_vector_type(2)));
typedef float v8f __attribute__((ext_vector_type(8)));
typedef int   v4i __attribute__((ext_vector_type(4)));

#define D_IN 256
#define D_HID 256
#define D_OUT 64

#define AS1 __attribute__((address_space(1)))
#define AS3 __attribute__((address_space(3)))

#if defined(__has_builtin)
#if __has_builtin(__builtin_amdgcn_global_load_async_to_lds_b128)
#define HAVE_ASYNC_LDS 1
#endif
#if __has_builtin(__builtin_amdgcn_s_wait_asynccnt)
#define WAIT_ASYNC() __builtin_amdgcn_s_wait_asynccnt(0)
#endif
#endif
#ifndef WAIT_ASYNC
#define WAIT_ASYNC() asm volatile("s_wait_asynccnt 0" ::: "memory")
#endif

// ---------------------------------------------------------------- zero fill
__global__ void zero_f32_kernel(float* __restrict__ p, long n) {
    long i = (long)blockIdx.x * blockDim.x + threadIdx.x;
    if (i < n) p[i] = 0.0f;
}

// ---------------------------------------------------------------- degrees
__global__ void degree_kernel(const int* __restrict__ src,
                              const int* __restrict__ dst,
                              float* __restrict__ outdeg,
                              float* __restrict__ indeg, int ne) {
    int e = blockIdx.x * blockDim.x + threadIdx.x;
    if (e < ne) {
        unsafeAtomicAdd(&outdeg[src[e]], 1.0f);   // non-returning f32 atomic add
        unsafeAtomicAdd(&indeg[dst[e]], 1.0f);
    }
}

// deg -> clamp(deg,1)^(-1/2), in place (works on concatenated outdeg|indeg)
__global__ void norm_kernel(float* __restrict__ deg, long n) {
    long i = (long)blockIdx.x * blockDim.x + threadIdx.x;
    if (i < n) {
        float d = fmaxf(deg[i], 1.0f);
        deg[i] = rsqrtf(d);
    }
}

// ------------------------------------------------- fused scale+gather+scatter
// agg[dst] += x[src] * out_norm[src]; one thread per (edge, 4-channel group),
// C = 256 channels fixed (both layers).
__global__ void scatter_kernel(const float* __restrict__ x,
                               const float* __restrict__ out_norm,
                               const int* __restrict__ src,
                               const int* __restrict__ dst,
                               float* __restrict__ agg, long ne) {
    long t = (long)blockIdx.x * blockDim.x + threadIdx.x;
    long total = ne << 6;                 // ne * 64 groups
    if (t >= total) return;
    int e  = (int)(t >> 6);
    int c4 = (int)(t & 63) << 2;
    int s = src[e];
    int d = dst[e];
    float nrm = out_norm[s];
    const float4 v = *reinterpret_cast<const float4*>(x + (long)s * 256 + c4);
    float* base = agg + (long)d * 256 + c4;
    unsafeAtomicAdd(base + 0, v.x * nrm);
    unsafeAtomicAdd(base + 1, v.y * nrm);
    unsafeAtomicAdd(base + 2, v.z * nrm);
    unsafeAtomicAdd(base + 3, v.w * nrm);
}

// ------------------------------------------------------------- WMMA GEMM
// Out[m, n] = act( in_norm[m] * sum_k A[m,k] * W[k,n] + bias[n] )
// Block = 256 threads (8 waves) covers 128 rows x 64 cols.
//  - stage W panel [256 x 64] in LDS via async global->LDS copies
//  - each wave: 16 rows x 64 cols = 4 x (16x16) accumulators,
//    K-loop of v_wmma_f32_16x16x4_f32; one A v2f load feeds 4 WMMAs.
// Fragment layouts per ISA 7.12.2:
//  A (16x4 f32): lanes 0-15 M=lane, VGPR0/1 = K 0/1; lanes 16-31 K=2/3.
//  B (4x16):     lanes 0-15 N=lane, K mirrored by lane half.
//  C/D:          VGPR r holds M=r (lanes 0-15) / M=r+8 (lanes 16-31), N=lane&15.
template <int KDIM, bool RELU>
__global__ void wmma_gemm_kernel(const float* __restrict__ A,
                                 const float* __restrict__ W,
                                 const float* __restrict__ bias,
                                 const float* __restrict__ in_norm,
                                 float* __restrict__ Out,
                                 int M, int NOUT) {
    __shared__ float bsh[KDIM * 64];           // 64 KB weight panel

    const int tid   = threadIdx.x;
    const int np    = blockIdx.x;              // 64-col N panel
    const int mbase = blockIdx.y * 128;

    // ---- stage B panel: KDIM*64 floats = KDIM*16 float4 units, 256 threads
    {
        const float* wpanel = W + np * 64;
        #pragma unroll
        for (int i = 0; i < (KDIM * 16) / 256; ++i) {
            int u   = tid + i * 256;           // float4 unit id
            int row = u >> 4;
            int c4  = (u & 15) << 2;
            const float* g = wpanel + (long)row * NOUT + c4;
            float* l = bsh + row * 64 + c4;
#ifdef HAVE_ASYNC_LDS
            __builtin_amdgcn_global_load_async_to_lds_b128(
                (AS1 v4i*)g, (AS3 v4i*)l, 0, 0);
#else
            *reinterpret_cast<float4*>(l) = *reinterpret_cast<const float4*>(g);
#endif
        }
#ifdef HAVE_ASYNC_LDS
        WAIT_ASYNC();
#endif
    }
    __syncthreads();

    const int wid   = tid >> 5;
    const int lane  = tid & 31;
    const int mrow0 = mbase + wid * 16;
    if (mrow0 >= M) return;                    // wave-uniform (after barrier)

    v8f acc0 = {}, acc1 = {}, acc2 = {}, acc3 = {};

    const int rowA  = mrow0 + (lane & 15);
    const int colL  = lane & 15;
    const int khalf = (lane >> 4) << 1;        // 0 or 2

    const float* aptr = A + (long)rowA * KDIM + khalf;
    const float* bl   = bsh + khalf * 64 + colL;

    #pragma unroll 2
    for (int k = 0; k < KDIM; k += 4) {
        v2f a = *reinterpret_cast<const v2f*>(aptr);   // K,K+1 contiguous (b64)
        v2f b0, b1, b2, b3;
        b0.x = bl[0];      b0.y = bl[64];
        b1.x = bl[16];     b1.y = bl[16 + 64];
        b2.x = bl[32];     b2.y = bl[32 + 64];
        b3.x = bl[48];     b3.y = bl[48 + 64];
        acc0 = __builtin_amdgcn_wmma_f32_16x16x4_f32(false, a, false, b0, (short)0, acc0, false, false);
        acc1 = __builtin_amdgcn_wmma_f32_16x16x4_f32(false, a, false, b1, (short)0, acc1, false, false);
        acc2 = __builtin_amdgcn_wmma_f32_16x16x4_f32(false, a, false, b2, (short)0, acc2, false, false);
        acc3 = __builtin_amdgcn_wmma_f32_16x16x4_f32(false, a, false, b3, (short)0, acc3, false, false);
        aptr += 4;
        bl   += 4 * 64;
    }

    // Epilogue: fuse in_norm row-scale, bias, relu; C/D layout mapping.
    const int rbase = mrow0 + ((lane >> 4) << 3);      // +8 for upper lane half
    float rs[8];
    #pragma unroll
    for (int r = 0; r < 8; ++r) rs[r] = in_norm[rbase + r];

    v8f accs[4] = {acc0, acc1, acc2, acc3};
    #pragma unroll
    for (int j = 0; j < 4; ++j) {
        const int col  = np * 64 + j * 16 + colL;
        const float bc = bias[col];
        #pragma unroll
        for (int r = 0; r < 8; ++r) {
            float v = accs[j][r] * rs[r] + bc;
            if (RELU) v = fmaxf(v, 0.0f);
            Out[(long)(rbase + r) * NOUT + col] = v;
        }
    }
}

// ---------------------------------------------------------------------------
extern "C" void kernel_launch(void* const* d_in, const int* in_sizes, int n_in,
                              void* d_out, int out_size, void* d_ws, size_t ws_size,
                              hipStream_t stream) {
    const float* features = (const float*)d_in[0];  // [NN, 256]
    const float* W1       = (const float*)d_in[1];  // [256, 256]
    const float* b1       = (const float*)d_in[2];  // [256]
    const float* W2       = (const float*)d_in[3];  // [256, 64]
    const float* b2       = (const float*)d_in[4];  // [64]
    const int*   src      = (const int*)d_in[5];    // [NE]
    const int*   dst      = (const int*)d_in[6];    // [NE]

    const long NN = in_sizes[0] / D_IN;   // 100000
    const long NE = in_sizes[5];          // 3200000

    // Workspace layout (floats): out_norm | in_norm | agg (reused) | h
    float* ws       = (float*)d_ws;
    float* out_norm = ws;
    float* in_norm  = ws + NN;
    float* agg      = ws + 2 * NN;
    float* h        = agg + NN * D_HID;

    const int B = 256;
    const long nAgg = NN * D_HID;

    // 1) zero degree buffers + accumulator
    {
        long n = 2 * NN;
        zero_f32_kernel<<<(n + B - 1) / B, B, 0, stream>>>(out_norm, n);
        zero_f32_kernel<<<(nAgg + B - 1) / B, B, 0, stream>>>(agg, nAgg);
    }
    // 2) degrees then norms (clamp>=1, rsqrt) over both arrays at once
    degree_kernel<<<(NE + B - 1) / B, B, 0, stream>>>(src, dst, out_norm, in_norm, (int)NE);
    {
        long n = 2 * NN;
        norm_kernel<<<(n + B - 1) / B, B, 0, stream>>>(out_norm, n);
    }

    // 3) layer 1: scatter (features * out_norm) into agg, then WMMA GEMM
    {
        long total = NE * 64;
        scatter_kernel<<<(total + B - 1) / B, B, 0, stream>>>(
            features, out_norm, src, dst, agg, NE);
    }
    {
        dim3 grid(D_HID / 64, (unsigned)((NN + 127) / 128));
        wmma_gemm_kernel<D_IN, true><<<grid, 256, 0, stream>>>(
            agg, W1, b1, in_norm, h, (int)NN, D_HID);
    }

    // 4) layer 2: re-zero accumulator, scatter (h * out_norm), WMMA GEMM -> d_out
    zero_f32_kernel<<<(nAgg + B - 1) / B, B, 0, stream>>>(agg, nAgg);
    {
        long total = NE * 64;
        scatter_kernel<<<(total + B - 1) / B, B, 0, stream>>>(
            h, out_norm, src, dst, agg, NE);
    }
    {
        dim3 grid(D_OUT / 64, (unsigned)((NN + 127) / 128));
        wmma_gemm_kernel<D_HID, false><<<grid, 256, 0, stream>>>(
            agg, W2, b2, in_norm, (float*)d_out, (int)NN, D_OUT);
    }
}